// SessionGraph_40072044872135
// MI455X (gfx1250) — compile-verified
//
#include <hip/hip_runtime.h>
#include <math.h>

// ---------------------------------------------------------------------------
// Shapes (fixed by the reference):
//   B=1024, S=50, H=100, H2=200, N_NODE=43098, NEI_N=5
// ---------------------------------------------------------------------------
#define BB   1024
#define SS   50
#define HH   100
#define H2   200
#define NOUT 43097

typedef __attribute__((ext_vector_type(2))) float v2f;
typedef __attribute__((ext_vector_type(8))) float v8f;
typedef __attribute__((ext_vector_type(4))) unsigned int u32x4;
typedef __attribute__((ext_vector_type(4))) int          i32x4;
typedef __attribute__((ext_vector_type(8))) int          i32x8;

#if defined(__HIP_DEVICE_COMPILE__) && __has_builtin(__builtin_amdgcn_tensor_load_to_lds)
#define HAVE_TDM 1
#else
#define HAVE_TDM 0
#endif

// ---------------------------------------------------------------------------
// Tensor Data Mover: issue a 2D tile load (global -> LDS) via D# descriptor.
// D# packing per CDNA5 ISA:
//   group0: [1:0]=count=1, [63:32]=lds_addr, [120:64]=global_addr, [127:126]=2
//   group1: data_size=2(4B), pad_enable, pad_interval/amount, tensor_dim0/1,
//           tile_dim0/1, tensor_dim0_stride (all element units)
// OOB reads (beyond tensor_dim) return zero -> free edge padding.
// LDS padding: pad_interval code (2<<c DWORDs), pad_amount code (c+1 DWORDs).
// ---------------------------------------------------------------------------
#if HAVE_TDM
__device__ __forceinline__
void tdm_load_2d(const void* gptr, unsigned lds_byte_off,
                 unsigned td0, unsigned td1,          // valid extents (elements)
                 unsigned tile0, unsigned tile1,      // tile dims (elements)
                 unsigned long long stride0,          // row stride (elements)
                 unsigned pad_interval_code, unsigned pad_amount_code)
{
    unsigned long long ga = (unsigned long long)(uintptr_t)gptr;
    u32x4 g0;
    g0[0] = 1u;                                          // count=1, user mode
    g0[1] = lds_byte_off;                                // lds_addr
    g0[2] = (unsigned)(ga & 0xffffffffull);              // global_addr[31:0]
    g0[3] = (unsigned)((ga >> 32) & 0x01ffffffu)         // global_addr[56:32]
          | (2u << 30);                                  // type=2 ("image")
    i32x8 g1;
    g1[0] = (int)((2u << 16)                             // data_size=4B
                | (1u << 20)                             // pad_enable
                | (pad_interval_code << 22)
                | (pad_amount_code << 25));
    g1[1] = (int)((td0 & 0xffffu) << 16);                // tensor_dim0[15:0]
    g1[2] = (int)((td0 >> 16) | ((td1 & 0xffffu) << 16));// td0[31:16]|td1[15:0]
    g1[3] = (int)((td1 >> 16) | ((tile0 & 0xffffu) << 16));
    g1[4] = (int)(tile1 & 0xffffu);                      // tile_dim1 (tile_dim2=0)
    g1[5] = (int)(unsigned)(stride0 & 0xffffffffull);    // dim0_stride[31:0]
    g1[6] = (int)(unsigned)((stride0 >> 32) & 0xffffu);  // dim0_stride[47:32]
    g1[7] = 0;
    i32x4 z4 = {};
#if defined(__clang_major__) && __clang_major__ >= 23
    i32x8 z8 = {};
    __builtin_amdgcn_tensor_load_to_lds(g0, g1, z4, z4, z8, 0);
#else
    __builtin_amdgcn_tensor_load_to_lds(g0, g1, z4, z4, 0);
#endif
}
#endif

// ---------------------------------------------------------------------------
// Generic tiled GEMM on the CDNA5 matrix pipe: C = A(MxK) * B(KxN) [+bias][relu]
// fp32 in / fp32 out via V_WMMA_F32_16X16X4_F32 (exact fp32 semantics, RNE).
// Block tile 64x64, K-slab 16 (4 wmma steps), 256 threads = 8 waves,
// each wave owns a 16x32 strip (two 16x16 accumulators).
// Tile staging by the Tensor Data Mover, double-buffered: wave 0 DMAs slab
// k+1 while all waves compute slab k (TENSORcnt in-order => wait<=2 keeps one
// tile-pair in flight).
// flags: bit0 = relu, bit1 = B is accessed transposed (B[n*ldb + k]).
// Batched via blockIdx.z with element strides sA/sB/sC.
// ---------------------------------------------------------------------------
#define TM 64
#define TN 64
#define TK 16

__global__ __launch_bounds__(256)
void gemm_wmma_f32(const float* __restrict__ A, const float* __restrict__ B,
                   const float* __restrict__ bias, float* __restrict__ C,
                   int M, int N, int K, int lda, int ldb, int ldc,
                   long long sA, long long sB, long long sC, int flags)
{
    const bool btr = (flags & 2) != 0;
    A += (long long)blockIdx.z * sA;
    B += (long long)blockIdx.z * sB;
    C += (long long)blockIdx.z * sC;

    __shared__ float As [2][TM][TK + 4];   // A slab, row stride 20 DW
    __shared__ float Bs [2][TK][TN + 4];   // B slab (row-major), row stride 68 DW
    __shared__ float Bst[2][TN][TK + 4];   // B slab (transposed), row stride 20 DW

    const int tid      = threadIdx.x;
    const int lane     = tid & 31;
    const int wave     = tid >> 5;      // 0..7
    const int laneIdx  = lane & 15;     // M (A) / N (B,D) index
    const int laneHalf = lane >> 4;     // K-pair select (A,B) / row+8 (D)
    const int m0       = (wave & 3) * 16;   // wave row within block tile
    const int n0       = (wave >> 2) * 32;  // wave col within block tile
    const int gm       = blockIdx.y * TM;
    const int gn       = blockIdx.x * TN;

    v8f acc0 = {};
    v8f acc1 = {};

#if HAVE_TDM
    // ---- TDM staging path -------------------------------------------------
    auto issue = [&](int kk0, int pp) {
        // A tile: TM x TK at (gm, kk0); OOB rows/cols zero-filled by TDM
        tdm_load_2d(A + (long long)gm * lda + kk0,
                    (unsigned)(uintptr_t)&As[pp][0][0],
                    (unsigned)(K - kk0), (unsigned)(M - gm),
                    TK, TM, (unsigned long long)lda,
                    /*16 DW rows*/3, /*pad 4 DW*/3);
        if (!btr) {
            // B tile: TK x TN at (kk0, gn)
            tdm_load_2d(B + (long long)kk0 * ldb + gn,
                        (unsigned)(uintptr_t)&Bs[pp][0][0],
                        (unsigned)(N - gn), (unsigned)(K - kk0),
                        TN, TK, (unsigned long long)ldb,
                        /*64 DW rows*/5, /*pad 4 DW*/3);
        } else {
            // B^T tile: TN x TK at (gn, kk0) -> lands N-major in LDS
            tdm_load_2d(B + (long long)gn * ldb + kk0,
                        (unsigned)(uintptr_t)&Bst[pp][0][0],
                        (unsigned)(K - kk0), (unsigned)(N - gn),
                        TK, TN, (unsigned long long)ldb,
                        /*16 DW rows*/3, /*pad 4 DW*/3);
        }
    };

    if (wave == 0) issue(0, 0);

    int p = 0;
    for (int k0 = 0; k0 < K; k0 += TK, p ^= 1) {
        if (wave == 0) {
            if (k0 + TK < K) {
                issue(k0 + TK, p ^ 1);                    // prefetch next slab
                __builtin_amdgcn_s_wait_tensorcnt(2);     // current pair done
            } else {
                __builtin_amdgcn_s_wait_tensorcnt(0);
            }
        }
        __syncthreads();

#pragma unroll
        for (int kk = 0; kk < TK; kk += 4) {
            const int ka = kk + 2 * laneHalf;
            v2f a;  a.x  = As[p][m0 + laneIdx][ka];
                    a.y  = As[p][m0 + laneIdx][ka + 1];
            v2f b0, b1;
            if (!btr) {
                b0.x = Bs[p][ka][n0 + laneIdx];      b0.y = Bs[p][ka + 1][n0 + laneIdx];
                b1.x = Bs[p][ka][n0 + 16 + laneIdx]; b1.y = Bs[p][ka + 1][n0 + 16 + laneIdx];
            } else {
                b0.x = Bst[p][n0 + laneIdx][ka];      b0.y = Bst[p][n0 + laneIdx][ka + 1];
                b1.x = Bst[p][n0 + 16 + laneIdx][ka]; b1.y = Bst[p][n0 + 16 + laneIdx][ka + 1];
            }
            acc0 = __builtin_amdgcn_wmma_f32_16x16x4_f32(false, a, false, b0,
                                                         (short)0, acc0, false, false);
            acc1 = __builtin_amdgcn_wmma_f32_16x16x4_f32(false, a, false, b1,
                                                         (short)0, acc1, false, false);
        }
        __syncthreads();   // all waves done reading buf p before it is reloaded
    }
#else
    // ---- Fallback: manual staging (no TDM builtin on this toolchain) ------
    for (int k0 = 0; k0 < K; k0 += TK) {
        for (int e = tid; e < TM * TK; e += 256) {
            int r = e / TK, c = e - r * TK;
            int gr = gm + r, gc = k0 + c;
            As[0][r][c] = (gr < M && gc < K) ? A[(long long)gr * lda + gc] : 0.0f;
        }
        if (!btr) {
            for (int e = tid; e < TK * TN; e += 256) {
                int r = e / TN, c = e - r * TN;
                int gk = k0 + r, gc = gn + c;
                Bs[0][r][c] = (gk < K && gc < N) ? B[(long long)gk * ldb + gc] : 0.0f;
            }
        } else {
            for (int e = tid; e < TK * TN; e += 256) {
                int c = e / TK, r = e - c * TK;
                int gk = k0 + r, gc = gn + c;
                Bst[0][c][r] = (gk < K && gc < N) ? B[(long long)gc * ldb + gk] : 0.0f;
            }
        }
        __syncthreads();
#pragma unroll
        for (int kk = 0; kk < TK; kk += 4) {
            const int ka = kk + 2 * laneHalf;
            v2f a;  a.x  = As[0][m0 + laneIdx][ka];
                    a.y  = As[0][m0 + laneIdx][ka + 1];
            v2f b0, b1;
            if (!btr) {
                b0.x = Bs[0][ka][n0 + laneIdx];      b0.y = Bs[0][ka + 1][n0 + laneIdx];
                b1.x = Bs[0][ka][n0 + 16 + laneIdx]; b1.y = Bs[0][ka + 1][n0 + 16 + laneIdx];
            } else {
                b0.x = Bst[0][n0 + laneIdx][ka];      b0.y = Bst[0][n0 + laneIdx][ka + 1];
                b1.x = Bst[0][n0 + 16 + laneIdx][ka]; b1.y = Bst[0][n0 + 16 + laneIdx][ka + 1];
            }
            acc0 = __builtin_amdgcn_wmma_f32_16x16x4_f32(false, a, false, b0,
                                                         (short)0, acc0, false, false);
            acc1 = __builtin_amdgcn_wmma_f32_16x16x4_f32(false, a, false, b1,
                                                         (short)0, acc1, false, false);
        }
        __syncthreads();
    }
#endif

    const bool relu = (flags & 1) != 0;
#pragma unroll
    for (int i = 0; i < 8; ++i) {
        int row = gm + m0 + laneHalf * 8 + i;   // D layout: VGPR i -> M = i (+8 hi lanes)
        if (row >= M) continue;
        int col0 = gn + n0 + laneIdx;
        if (col0 < N) {
            float v = acc0[i];
            if (bias) v += bias[col0];
            if (relu) v = v > 0.0f ? v : 0.0f;
            C[(long long)row * ldc + col0] = v;
        }
        int col1 = col0 + 16;
        if (col1 < N) {
            float v = acc1[i];
            if (bias) v += bias[col1];
            if (relu) v = v > 0.0f ? v : 0.0f;
            C[(long long)row * ldc + col1] = v;
        }
    }
}

// ---------------------------------------------------------------------------
// X[b,s,:] = concat(embedding[items[b,s]], pos_emb[s])   -> (B*S, 200)
// ---------------------------------------------------------------------------
__global__ void k_build_x(const int* __restrict__ items, const float* __restrict__ emb,
                          const float* __restrict__ pos, float* __restrict__ X, int total)
{
    int t = blockIdx.x * blockDim.x + threadIdx.x;
    if (t >= total) return;
    int row = t / H2;
    int c   = t - row * H2;
    int s   = row % SS;
    X[t] = (c < HH) ? emb[(long long)items[row] * HH + c]
                    : pos[s * HH + (c - HH)];
}

// ---------------------------------------------------------------------------
// ht[b] = hidden[b, last_idx(b)],  last_idx = (int)sum(mask[b]) - 1
// ---------------------------------------------------------------------------
__global__ __launch_bounds__(128)
void k_ht(const float* __restrict__ hidden, const float* __restrict__ mask,
          float* __restrict__ ht)
{
    int b = blockIdx.x, h = threadIdx.x;
    float sm = 0.0f;
    for (int s = 0; s < SS; ++s) sm += mask[b * SS + s];
    int last = (int)sm - 1;
    last = last < 0 ? 0 : (last >= SS ? SS - 1 : last);
    if (h < HH)
        ht[b * HH + h] = hidden[((long long)b * SS + last) * HH + h];
}

// ---------------------------------------------------------------------------
// alpha[b,s] = sigmoid(q1[b]+q2[b,s]) . w3 ;  a[b] = sum_s alpha*mask*hidden
// ---------------------------------------------------------------------------
__global__ __launch_bounds__(128)
void k_attn(const float* __restrict__ q1, const float* __restrict__ q2,
            const float* __restrict__ hidden, const float* __restrict__ mask,
            const float* __restrict__ w3, float* __restrict__ a)
{
    __shared__ float red[128];
    int b = blockIdx.x, h = threadIdx.x;
    bool act = h < HH;
    float q1v = act ? q1[b * HH + h] : 0.0f;
    float w3v = act ? w3[h] : 0.0f;
    float acc = 0.0f;
    for (int s = 0; s < SS; ++s) {
        float q2v = act ? q2[((long long)b * SS + s) * HH + h] : 0.0f;
        float sig = 1.0f / (1.0f + expf(-(q1v + q2v)));
        red[h] = act ? sig * w3v : 0.0f;
        __syncthreads();
        for (int st = 64; st > 0; st >>= 1) {
            if (h < st) red[h] += red[h + st];
            __syncthreads();
        }
        float alpha = red[0];
        __syncthreads();
        if (act) acc += alpha * mask[b * SS + s] * hidden[((long long)b * SS + s) * HH + h];
    }
    if (act) a[b * HH + h] = acc;
}

// ---------------------------------------------------------------------------
// cat[b] = [a[b] | ht[b]]   -> (B, 200)
// ---------------------------------------------------------------------------
__global__ void k_cat(const float* __restrict__ a, const float* __restrict__ ht,
                      float* __restrict__ cat, int total)
{
    int t = blockIdx.x * blockDim.x + threadIdx.x;
    if (t >= total) return;
    int b = t / H2, c = t - b * H2;
    cat[t] = (c < HH) ? a[b * HH + c] : ht[b * HH + (c - HH)];
}

// ---------------------------------------------------------------------------
// Per-row: cos = fenzi / (ni*nj); softmax(1024); top-5 (tie -> lower index);
// softmax(top-5); neighbor = sum w_t * sess[idx_t]; sessf = sess + neighbor.
// Norms recovered from the Gram diagonal: ni = sqrt(fenzi[i,i] + 100*1e-6).
// ---------------------------------------------------------------------------
__global__ __launch_bounds__(256)
void k_neighbor(const float* __restrict__ fenzi, const float* __restrict__ sess,
                float* __restrict__ sessf)
{
    __shared__ float cs[BB];
    __shared__ float red[256];
    __shared__ int   redi[256];
    __shared__ float topw[5];
    __shared__ int   topi[5];

    int i = blockIdx.x, tid = threadIdx.x;
    float ni = sqrtf(fenzi[(long long)i * BB + i] + 1e-4f);
    for (int j = tid; j < BB; j += 256) {
        float nj = sqrtf(fenzi[(long long)j * BB + j] + 1e-4f);
        cs[j] = fenzi[(long long)i * BB + j] / (ni * nj);
    }
    __syncthreads();

    float mx = -1e30f;
    for (int j = tid; j < BB; j += 256) mx = fmaxf(mx, cs[j]);
    red[tid] = mx; __syncthreads();
    for (int st = 128; st > 0; st >>= 1) {
        if (tid < st) red[tid] = fmaxf(red[tid], red[tid + st]);
        __syncthreads();
    }
    mx = red[0]; __syncthreads();

    float sm = 0.0f;
    for (int j = tid; j < BB; j += 256) { float e = expf(cs[j] - mx); cs[j] = e; sm += e; }
    red[tid] = sm; __syncthreads();
    for (int st = 128; st > 0; st >>= 1) {
        if (tid < st) red[tid] += red[tid + st];
        __syncthreads();
    }
    float Z = red[0]; __syncthreads();
    for (int j = tid; j < BB; j += 256) cs[j] /= Z;
    __syncthreads();

    for (int t = 0; t < 5; ++t) {
        float bv = -1e30f; int bi = BB;
        for (int j = tid; j < BB; j += 256) {
            float v = cs[j];
            if (v > bv || (v == bv && j < bi)) { bv = v; bi = j; }
        }
        red[tid] = bv; redi[tid] = bi; __syncthreads();
        for (int st = 128; st > 0; st >>= 1) {
            if (tid < st) {
                if (red[tid + st] > red[tid] ||
                    (red[tid + st] == red[tid] && redi[tid + st] < redi[tid])) {
                    red[tid] = red[tid + st]; redi[tid] = redi[tid + st];
                }
            }
            __syncthreads();
        }
        if (tid == 0) { topw[t] = red[0]; topi[t] = redi[0]; cs[redi[0]] = -1e30f; }
        __syncthreads();
    }

    if (tid == 0) {
        float m2 = topw[0], z2 = 0.0f, e[5];
        for (int t = 0; t < 5; ++t) { e[t] = expf(topw[t] - m2); z2 += e[t]; }
        for (int t = 0; t < 5; ++t) topw[t] = e[t] / z2;
    }
    __syncthreads();

    if (tid < HH) {
        float nb = 0.0f;
        for (int t = 0; t < 5; ++t) nb += topw[t] * sess[(long long)topi[t] * HH + tid];
        sessf[i * HH + tid] = sess[i * HH + tid] + nb;
    }
}

// ---------------------------------------------------------------------------
// Orchestration. Workspace layout (floats):
//   [0, 10.24M)       region R0  (X -> S1 -> hidden)
//   [10.24M, 20.48M)  region R1  (T1 -> T2 -> q2)
//   then fenzi(1M), ht, q1, a, cat, sess, sessf   (total ~89 MB)
// ---------------------------------------------------------------------------
extern "C" void kernel_launch(void* const* d_in, const int* in_sizes, int n_in,
                              void* d_out, int out_size, void* d_ws, size_t ws_size,
                              hipStream_t stream)
{
    (void)in_sizes; (void)n_in; (void)out_size; (void)ws_size;

    const int*   items = (const int*)  d_in[0];
    const float* adj   = (const float*)d_in[1];
    const float* mask  = (const float*)d_in[2];
    const float* emb   = (const float*)d_in[3];
    const float* pos   = (const float*)d_in[4];
    const float* Wgc1  = (const float*)d_in[5];
    const float* bgc1  = (const float*)d_in[6];
    const float* Wgc2  = (const float*)d_in[7];
    const float* bgc2  = (const float*)d_in[8];
    const float* W1    = (const float*)d_in[9];
    const float* b1    = (const float*)d_in[10];
    const float* W2    = (const float*)d_in[11];
    const float* b2    = (const float*)d_in[12];
    const float* w3    = (const float*)d_in[13];
    const float* Wt    = (const float*)d_in[14];
    const float* bt    = (const float*)d_in[15];
    float* out = (float*)d_out;

    float* W     = (float*)d_ws;
    float* R0    = W;                       // 10,240,000 floats
    float* R1    = W + 10240000;            // 10,240,000 floats
    float* fenzi = W + 20480000;            // 1,048,576
    float* ht    = fenzi + 1048576;         // 102,400
    float* q1    = ht    + 102400;          // 102,400
    float* av    = q1    + 102400;          // 102,400
    float* cat   = av    + 102400;          // 204,800
    float* sess  = cat   + 204800;          // 102,400
    float* sessf = sess  + 102400;          // 102,400

    const int MS = BB * SS; // 51200

    // X = [emb[items] | pos]  -> R0
    k_build_x<<<(MS * H2 + 255) / 256, 256, 0, stream>>>(items, emb, pos, R0, MS * H2);

    // T1 = X @ Wgc1 + bgc1                         (51200 x 200 x 200) -> R1
    gemm_wmma_f32<<<dim3(4, 800, 1), 256, 0, stream>>>(
        R0, Wgc1, bgc1, R1, MS, H2, H2, H2, H2, H2, 0, 0, 0, 0);

    // S1 = relu(adj @ T1)  batched                 (1024 x [50 x 200 x 50]) -> R0
    gemm_wmma_f32<<<dim3(4, 1, BB), 256, 0, stream>>>(
        adj, R1, nullptr, R0, SS, H2, SS, SS, H2, H2,
        (long long)SS * SS, (long long)SS * H2, (long long)SS * H2, 1);

    // T2 = S1 @ Wgc2[:, :100] + bgc2[:100]         (51200 x 100 x 200) -> R1
    gemm_wmma_f32<<<dim3(2, 800, 1), 256, 0, stream>>>(
        R0, Wgc2, bgc2, R1, MS, HH, H2, H2, H2, HH, 0, 0, 0, 0);

    // hidden = adj @ T2  batched                   (1024 x [50 x 100 x 50]) -> R0
    gemm_wmma_f32<<<dim3(2, 1, BB), 256, 0, stream>>>(
        adj, R1, nullptr, R0, SS, HH, SS, SS, HH, HH,
        (long long)SS * SS, (long long)SS * HH, (long long)SS * HH, 0);
    float* hidden = R0;

    // ht[b] = hidden[b, last_idx]
    k_ht<<<BB, 128, 0, stream>>>(hidden, mask, ht);

    // q1 = ht @ W1 + b1                            (1024 x 100 x 100)
    gemm_wmma_f32<<<dim3(2, 16, 1), 256, 0, stream>>>(
        ht, W1, b1, q1, BB, HH, HH, HH, HH, HH, 0, 0, 0, 0);

    // q2 = hidden @ W2 + b2                        (51200 x 100 x 100) -> R1
    gemm_wmma_f32<<<dim3(2, 800, 1), 256, 0, stream>>>(
        hidden, W2, b2, R1, MS, HH, HH, HH, HH, HH, 0, 0, 0, 0);

    // a = sum_s sigmoid(q1+q2).w3 * mask * hidden
    k_attn<<<BB, 128, 0, stream>>>(q1, R1, hidden, mask, w3, av);

    // cat = [a | ht]
    k_cat<<<(BB * H2 + 255) / 256, 256, 0, stream>>>(av, ht, cat, BB * H2);

    // sess = cat @ Wt + bt                         (1024 x 100 x 200)
    gemm_wmma_f32<<<dim3(2, 16, 1), 256, 0, stream>>>(
        cat, Wt, bt, sess, BB, HH, H2, H2, HH, HH, 0, 0, 0, 0);

    // fenzi = sess @ sess^T                        (1024 x 1024 x 100)
    gemm_wmma_f32<<<dim3(16, 16, 1), 256, 0, stream>>>(
        sess, sess, nullptr, fenzi, BB, BB, HH, HH, HH, BB, 0, 0, 0, 2);

    // neighbor mix -> sessf
    k_neighbor<<<BB, 256, 0, stream>>>(fenzi, sess, sessf);

    // scores = sessf @ embedding[1:].T             (1024 x 43097 x 100) -> out
    gemm_wmma_f32<<<dim3((NOUT + 63) / 64, 16, 1), 256, 0, stream>>>(
        sessf, emb + HH, nullptr, out, BB, NOUT, HH, HH, HH, NOUT, 0, 0, 0, 2);
}